// GNN_10539849744444
// MI455X (gfx1250) — compile-verified
//
#include <hip/hip_runtime.h>
#include <hip/hip_bf16.h>

// ---------------- problem constants (from reference) ----------------
#define N_NODES  100000
#define F_IN     1433
#define F_HID    32
#define F_OUT    7
#define N_EDGES  3200000
#define E_TOT    (N_EDGES + N_NODES)   // with self-loops
#define KP1      1436                  // F_IN rounded up to multiple of 4
#define CHUNK    256                   // K-rows of W1 staged in LDS at a time

typedef __attribute__((ext_vector_type(2))) float v2f;
typedef __attribute__((ext_vector_type(8))) float v8f;

// ---------------- degree / dinv ----------------
__global__ __launch_bounds__(256) void k_init_deg(float* deg) {
    int i = blockIdx.x * 256 + threadIdx.x;
    if (i < N_NODES) deg[i] = 1.0f;                 // self-loop contribution
}

__global__ __launch_bounds__(256) void k_deg_scatter(const int* __restrict__ dst,
                                                     float* __restrict__ deg) {
    int e = blockIdx.x * 256 + threadIdx.x;
    if (e < N_EDGES) atomicAdd(&deg[dst[e]], 1.0f);
}

__global__ __launch_bounds__(256) void k_rsqrt(float* d) {
    int i = blockIdx.x * 256 + threadIdx.x;
    if (i < N_NODES) d[i] = rsqrtf(d[i]);           // deg >= 1 always
}

// ---------------- GEMM1: H = X @ W1  (WMMA f32 16x16x4) ----------------
// block = 256 threads = 8 waves; wave w owns rows [blk*128 + w*16, +16),
// and both 16-wide column tiles of the 32-wide output.
__global__ __launch_bounds__(256) void k_gemm1_wmma(const float* __restrict__ X,
                                                    const float* __restrict__ W,
                                                    float* __restrict__ H) {
    __shared__ float lW[CHUNK * F_HID];             // 32 KB staging for W1

    const int tid  = threadIdx.x;
    const int lane = tid & 31;
    const int wave = tid >> 5;
    const int m    = lane & 15;
    const int koff = (lane >> 4) << 1;              // 0 for lanes 0-15, 2 for 16-31
    const int rowb = blockIdx.x * 128 + wave * 16;

    const int arow_i = rowb + m;
    const float* xrow = X + (size_t)((arow_i < N_NODES) ? arow_i : (N_NODES - 1)) * F_IN;

    v8f c0 = {0.f,0.f,0.f,0.f,0.f,0.f,0.f,0.f};
    v8f c1 = {0.f,0.f,0.f,0.f,0.f,0.f,0.f,0.f};

    for (int kc = 0; kc < KP1; kc += CHUNK) {
        __syncthreads();                            // protect LDS reuse
        for (int i = tid; i < CHUNK * F_HID; i += 256) {
            int kg = kc + (i >> 5);                 // K row (stride 32 cols)
            lW[i] = (kg < F_IN) ? W[kg * F_HID + (i & 31)] : 0.0f;
        }
        __syncthreads();

        const int kend = (KP1 - kc < CHUNK) ? (KP1 - kc) : CHUNK;
        if (kc + CHUNK <= F_IN) {
            // full chunk, no bounds checks on A
            for (int k = 0; k < CHUNK; k += 4) {
                if ((k & 63) == 0)
                    __builtin_prefetch(xrow + kc + k + 512, 0, 1);
                int kk = k + koff;
                v2f a; a.x = xrow[kc + kk]; a.y = xrow[kc + kk + 1];
                v2f b0; b0.x = lW[kk * F_HID + m];        b0.y = lW[(kk + 1) * F_HID + m];
                v2f b1; b1.x = lW[kk * F_HID + m + 16];   b1.y = lW[(kk + 1) * F_HID + m + 16];
                c0 = __builtin_amdgcn_wmma_f32_16x16x4_f32(false, a, false, b0, (short)0, c0, false, false);
                c1 = __builtin_amdgcn_wmma_f32_16x16x4_f32(false, a, false, b1, (short)0, c1, false, false);
            }
        } else {
            // tail chunk: A guarded (LDS already zero-padded)
            for (int k = 0; k < kend; k += 4) {
                int kk = k + koff;
                int kg = kc + kk;
                v2f a;
                a.x = (kg     < F_IN) ? xrow[kg]     : 0.0f;
                a.y = (kg + 1 < F_IN) ? xrow[kg + 1] : 0.0f;
                v2f b0; b0.x = lW[kk * F_HID + m];        b0.y = lW[(kk + 1) * F_HID + m];
                v2f b1; b1.x = lW[kk * F_HID + m + 16];   b1.y = lW[(kk + 1) * F_HID + m + 16];
                c0 = __builtin_amdgcn_wmma_f32_16x16x4_f32(false, a, false, b0, (short)0, c0, false, false);
                c1 = __builtin_amdgcn_wmma_f32_16x16x4_f32(false, a, false, b1, (short)0, c1, false, false);
            }
        }
    }

    // D layout: VGPR r -> row rowb + (lane>=16 ? r+8 : r), col = lane&15
    const int r0 = rowb + ((lane >> 4) << 3);
    #pragma unroll
    for (int r = 0; r < 8; ++r) {
        int rr = r0 + r;
        if (rr < N_NODES) {
            H[(size_t)rr * F_HID + m]      = c0[r];
            H[(size_t)rr * F_HID + m + 16] = c1[r];
        }
    }
}

// ---------------- init out = bias broadcast ----------------
__global__ __launch_bounds__(256) void k_init_out1(const float* __restrict__ b,
                                                   float* __restrict__ o) {
    int i = blockIdx.x * 256 + threadIdx.x;
    if (i < N_NODES * F_HID) o[i] = b[i & 31];
}

__global__ __launch_bounds__(256) void k_init_out2(const float* __restrict__ b,
                                                   float* __restrict__ o) {
    int i = blockIdx.x * 256 + threadIdx.x;
    if (i < N_NODES * F_OUT) o[i] = b[i % F_OUT];
}

// ---------------- edge scatter, layer 1 (32 features, 8 lanes/edge) ----------------
__global__ __launch_bounds__(256) void k_scatter1(const int* __restrict__ src,
                                                  const int* __restrict__ dst,
                                                  const float* __restrict__ dinv,
                                                  const float* __restrict__ h,
                                                  float* __restrict__ out) {
    int tid = blockIdx.x * 256 + threadIdx.x;      // < 26.4M
    int e = tid >> 3;
    if (e >= E_TOT) return;
    int g = (tid & 7) << 2;                        // feature group of 4
    int s, d;
    if (e < N_EDGES) { s = src[e]; d = dst[e]; }
    else             { s = d = e - N_EDGES; }      // self-loop
    float nrm = dinv[s] * dinv[d];
    const float4 hv = *reinterpret_cast<const float4*>(h + (size_t)s * F_HID + g);
    float* o = out + (size_t)d * F_HID + g;
    atomicAdd(o + 0, hv.x * nrm);
    atomicAdd(o + 1, hv.y * nrm);
    atomicAdd(o + 2, hv.z * nrm);
    atomicAdd(o + 3, hv.w * nrm);
}

// ---------------- ReLU in place ----------------
__global__ __launch_bounds__(256) void k_relu(float* o) {
    int i = blockIdx.x * 256 + threadIdx.x;
    if (i < N_NODES * F_HID) o[i] = fmaxf(o[i], 0.0f);
}

// ---------------- GEMM2: Z = relu(out1) @ W2  (WMMA, K=32, N padded to 16) ----------------
__global__ __launch_bounds__(256) void k_gemm2_wmma(const float* __restrict__ A,
                                                    const float* __restrict__ W2,
                                                    float* __restrict__ Z) {
    __shared__ float lW[F_HID * 16];               // 32 x 16, zero-padded cols 7..15

    const int tid  = threadIdx.x;
    const int lane = tid & 31;
    const int wave = tid >> 5;
    const int m    = lane & 15;
    const int koff = (lane >> 4) << 1;
    const int rowb = blockIdx.x * 128 + wave * 16;

    for (int i = tid; i < F_HID * 16; i += 256) {
        int k = i >> 4, col = i & 15;
        lW[i] = (col < F_OUT) ? W2[k * F_OUT + col] : 0.0f;
    }
    __syncthreads();

    const int arow_i = rowb + m;
    const float* arow = A + (size_t)((arow_i < N_NODES) ? arow_i : (N_NODES - 1)) * F_HID;

    v8f c = {0.f,0.f,0.f,0.f,0.f,0.f,0.f,0.f};
    #pragma unroll
    for (int k = 0; k < F_HID; k += 4) {
        int kk = k + koff;
        v2f a; a.x = arow[kk];            a.y = arow[kk + 1];
        v2f b; b.x = lW[kk * 16 + m];     b.y = lW[(kk + 1) * 16 + m];
        c = __builtin_amdgcn_wmma_f32_16x16x4_f32(false, a, false, b, (short)0, c, false, false);
    }

    if (m < F_OUT) {
        const int r0 = rowb + ((lane >> 4) << 3);
        #pragma unroll
        for (int r = 0; r < 8; ++r) {
            int rr = r0 + r;
            if (rr < N_NODES) Z[(size_t)rr * F_OUT + m] = c[r];
        }
    }
}

// ---------------- edge scatter, layer 2 (7 features, 8 lanes/edge) ----------------
__global__ __launch_bounds__(256) void k_scatter2(const int* __restrict__ src,
                                                  const int* __restrict__ dst,
                                                  const float* __restrict__ dinv,
                                                  const float* __restrict__ z,
                                                  float* __restrict__ out) {
    int tid = blockIdx.x * 256 + threadIdx.x;
    int e = tid >> 3;
    int f = tid & 7;
    if (e >= E_TOT || f >= F_OUT) return;
    int s, d;
    if (e < N_EDGES) { s = src[e]; d = dst[e]; }
    else             { s = d = e - N_EDGES; }
    float nrm = dinv[s] * dinv[d];
    atomicAdd(out + (size_t)d * F_OUT + f, z[(size_t)s * F_OUT + f] * nrm);
}

// ---------------- per-node log_softmax over 7 classes ----------------
__global__ __launch_bounds__(256) void k_logsoftmax(const float* __restrict__ z,
                                                    float* __restrict__ out) {
    int i = blockIdx.x * 256 + threadIdx.x;
    if (i >= N_NODES) return;
    const float* zr = z + (size_t)i * F_OUT;
    float v[F_OUT];
    float mx = -3.402823466e38f;
    #pragma unroll
    for (int f = 0; f < F_OUT; ++f) { v[f] = zr[f]; mx = fmaxf(mx, v[f]); }
    float s = 0.0f;
    #pragma unroll
    for (int f = 0; f < F_OUT; ++f) s += __expf(v[f] - mx);
    float l = mx + __logf(s);
    float* o = out + (size_t)i * F_OUT;
    #pragma unroll
    for (int f = 0; f < F_OUT; ++f) o[f] = v[f] - l;
}

// ---------------- launch ----------------
extern "C" void kernel_launch(void* const* d_in, const int* in_sizes, int n_in,
                              void* d_out, int out_size, void* d_ws, size_t ws_size,
                              hipStream_t stream) {
    const float* X  = (const float*)d_in[0];
    const int*   EI = (const int*)  d_in[1];
    const float* W1 = (const float*)d_in[2];
    const float* B1 = (const float*)d_in[3];
    const float* W2 = (const float*)d_in[4];
    const float* B2 = (const float*)d_in[5];
    float*       O  = (float*)d_out;

    const int* src = EI;
    const int* dst = EI + N_EDGES;

    // workspace partition (floats)
    float* ws   = (float*)d_ws;
    float* dinv = ws;                                   // N
    float* h1   = dinv + N_NODES;                       // N*32
    float* o1   = h1   + (size_t)N_NODES * F_HID;       // N*32
    float* z2   = o1   + (size_t)N_NODES * F_HID;       // N*7
    float* o2   = z2   + (size_t)N_NODES * F_OUT;       // N*7

    const int gN    = (N_NODES + 255) / 256;                 // 391
    const int gE    = (N_EDGES + 255) / 256;                 // 12500
    const int gNF   = (N_NODES * F_HID + 255) / 256;         // 12500
    const int gNO   = (N_NODES * F_OUT + 255) / 256;         // 2735
    const int gRows = (N_NODES + 127) / 128;                 // 782
    const int gEdg  = (E_TOT * 8 + 255) / 256;               // 103125

    // degree / normalization
    k_init_deg   <<<gN,   256, 0, stream>>>(dinv);
    k_deg_scatter<<<gE,   256, 0, stream>>>(dst, dinv);
    k_rsqrt      <<<gN,   256, 0, stream>>>(dinv);

    // layer 1
    k_gemm1_wmma <<<gRows, 256, 0, stream>>>(X, W1, h1);
    k_init_out1  <<<gNF,  256, 0, stream>>>(B1, o1);
    k_scatter1   <<<gEdg, 256, 0, stream>>>(src, dst, dinv, h1, o1);
    k_relu       <<<gNF,  256, 0, stream>>>(o1);

    // layer 2
    k_gemm2_wmma <<<gRows, 256, 0, stream>>>(o1, W2, z2);
    k_init_out2  <<<gNO,  256, 0, stream>>>(B2, o2);
    k_scatter2   <<<gEdg, 256, 0, stream>>>(src, dst, dinv, z2, o2);

    // output
    k_logsoftmax <<<gN,   256, 0, stream>>>(o2, O);
}